// MyLinear_89275190214836
// MI455X (gfx1250) — compile-verified
//
#include <hip/hip_runtime.h>

// Mitchell approximate GEMM on gfx1250:
//   mitchell(a,b) = a*b + sign*2^(ea+eb) * ( -ua*ub + relu(ua+ub-1) )
//   a' = a - sign(a)*2^ea  (bits & 0xFF800000)  =>  a'*b' = sign*2^(ea+eb)*ua*ub
//   relu cross-term LS fit: relu(ua+ub-1) ~= 0.825*ua*ub
//   =>  out ~= X*W^T - 0.175*X'*W'^T       (two dense f32 GEMMs -> v_wmma_f32_16x16x4_f32)
// Data movement: double-buffered TDM (tensor_load_to_lds, 6-arg builtin) stages 64x64 f32
// tiles of X and W into LDS; waves consume fragments via ds loads.
// Sync: s_wait_tensorcnt + workgroup barrier.

typedef __attribute__((ext_vector_type(2))) float    v2f;
typedef __attribute__((ext_vector_type(8))) float    v8f;
typedef __attribute__((ext_vector_type(4))) unsigned v4u;
typedef __attribute__((ext_vector_type(8))) int      v8i;
typedef __attribute__((ext_vector_type(4))) int      v4i;

#define CORR (-0.175f)
#define KC 64    // K elements staged per chunk
#define BT 64    // block tile edge (rows and cols)

__device__ __forceinline__ float mantissa_residue(float v) {
    unsigned u = __float_as_uint(v) & 0xFF800000u;   // sign(v)*2^floor(log2|v|)
    return v - __uint_as_float(u);
}

__device__ __forceinline__ v8f wmma_f32(v2f a, v2f b, v8f c) {
    return __builtin_amdgcn_wmma_f32_16x16x4_f32(false, a, false, b, (short)0, c,
                                                 false, false);
}

// TDM: load a BT x KC f32 tile (row-major, row stride = K elements) into LDS.
__device__ __forceinline__ void tdm_load_tile(const float* gsrc, unsigned lds_off, int K) {
    unsigned long long ga = (unsigned long long)(uintptr_t)gsrc;

    v4u g0;
    g0[0] = 1u;                                              // count=1 (valid, user mode)
    g0[1] = lds_off;                                         // LDS byte address
    g0[2] = (unsigned)(ga & 0xFFFFFFFFu);                    // global_addr[31:0]
    g0[3] = (unsigned)((ga >> 32) & 0x01FFFFFFu) | (2u << 30); // addr[56:32] | type=2

    const unsigned td0 = (unsigned)K;        // tensor dim0 = full row length
    const unsigned td1 = 1u << 20;           // generous dim1; tile is interior, no OOB
    const unsigned long long s0 = (unsigned long long)K;  // dim0 stride (elements)

    v8i g1;
    g1[0] = (int)(2u << 16);                                  // wg_mask=0 | data_size=4B
    g1[1] = (int)((td0 & 0xFFFFu) << 16);                     // barrier_addr=0 | dim0 lo
    g1[2] = (int)((td0 >> 16) | ((td1 & 0xFFFFu) << 16));     // dim0 hi | dim1 lo
    g1[3] = (int)((td1 >> 16) | ((unsigned)KC << 16));        // dim1 hi | tile_dim0=KC
    g1[4] = (int)(BT);                                        // tile_dim1=BT | tile_dim2=0
    g1[5] = (int)(s0 & 0xFFFFFFFFu);                          // dim0_stride lo
    g1[6] = (int)((s0 >> 32) & 0xFFFFu);                      // dim0_stride hi | dim1_stride=0
    g1[7] = 0;

    v4i z4 = {};   // groups 2/3 unused (2D tensor)
    v8i z8 = {};   // trailing group (clang-23 6-arg form)
    __builtin_amdgcn_tensor_load_to_lds(g0, g1, z4, z4, z8, 0);
}

__global__ __launch_bounds__(128)
void mitchell_linear_wmma_tdm(const float* __restrict__ x,
                              const float* __restrict__ w,
                              const float* __restrict__ bias,
                              float* __restrict__ out,
                              int K, int M) {
    __shared__ float xs[2][BT * KC];   // X tiles (rows r0..r0+63, KC k's)
    __shared__ float ws[2][BT * KC];   // W tiles (out-cols c0..c0+63, KC k's)

    const int lane = threadIdx.x & 31;
    const int wave = threadIdx.x >> 5;
    const int half = lane >> 4;        // 0: lanes 0-15, 1: lanes 16-31
    const int lc   = lane & 15;

    const int r0 = blockIdx.y * BT;
    const int c0 = blockIdx.x * BT;
    const int rw = (wave >> 1) * 32;   // wave's 32x32 sub-tile within the 64x64 block tile
    const int cw = (wave & 1) * 32;

    // LDS byte offsets (flat LDS aperture: low 32 bits = LDS offset)
    const unsigned xoff0 = (unsigned)(uintptr_t)(void*)&xs[0][0];
    const unsigned xoff1 = (unsigned)(uintptr_t)(void*)&xs[1][0];
    const unsigned woff0 = (unsigned)(uintptr_t)(void*)&ws[0][0];
    const unsigned woff1 = (unsigned)(uintptr_t)(void*)&ws[1][0];

    const int NC = K / KC;

    if (wave == 0) {   // prime the pipeline: chunks 0 and 1
        tdm_load_tile(x + (size_t)r0 * K, xoff0, K);
        tdm_load_tile(w + (size_t)c0 * K, woff0, K);
        tdm_load_tile(x + (size_t)r0 * K + KC, xoff1, K);
        tdm_load_tile(w + (size_t)c0 * K + KC, woff1, K);
    }

    v8f acc00 = {}; v8f acc01 = {}; v8f acc10 = {}; v8f acc11 = {};

    const int ka = 2 * half;   // A 16x4 layout: lanes0-15 K+{0,1}; lanes16-31 K+{2,3}
    const int kb = half;       // B 4x16 layout: VGPR0 rows {K,K+1}; VGPR1 rows {K+2,K+3}

    for (int c = 0; c < NC; ++c) {
        if (wave == 0) {       // TDM is in-order per wave: <=2 outstanding ==> chunk c landed
            if (c + 1 < NC) __builtin_amdgcn_s_wait_tensorcnt(2);
            else            __builtin_amdgcn_s_wait_tensorcnt(0);
        }
        __syncthreads();

        const float* xb = &xs[c & 1][0];
        const float* wb = &ws[c & 1][0];

        #pragma unroll 4
        for (int kk = 0; kk < KC; kk += 4) {
            v2f a0 = *(const v2f*)(xb + (rw + lc) * KC + kk + ka);
            v2f a1 = *(const v2f*)(xb + (rw + 16 + lc) * KC + kk + ka);
            v2f b0, b1;
            b0.x = wb[(cw + lc) * KC + kk + kb];
            b0.y = wb[(cw + lc) * KC + kk + kb + 2];
            b1.x = wb[(cw + 16 + lc) * KC + kk + kb];
            b1.y = wb[(cw + 16 + lc) * KC + kk + kb + 2];

            v2f a0c, a1c, b0c, b1c;   // log-residues; CORR folded into A side
            a0c.x = mantissa_residue(a0.x) * CORR;  a0c.y = mantissa_residue(a0.y) * CORR;
            a1c.x = mantissa_residue(a1.x) * CORR;  a1c.y = mantissa_residue(a1.y) * CORR;
            b0c.x = mantissa_residue(b0.x);         b0c.y = mantissa_residue(b0.y);
            b1c.x = mantissa_residue(b1.x);         b1c.y = mantissa_residue(b1.y);

            acc00 = wmma_f32(a0,  b0,  acc00);
            acc00 = wmma_f32(a0c, b0c, acc00);
            acc01 = wmma_f32(a0,  b1,  acc01);
            acc01 = wmma_f32(a0c, b1c, acc01);
            acc10 = wmma_f32(a1,  b0,  acc10);
            acc10 = wmma_f32(a1c, b0c, acc10);
            acc11 = wmma_f32(a1,  b1,  acc11);
            acc11 = wmma_f32(a1c, b1c, acc11);
        }

        __syncthreads();       // everyone done reading buf (c&1) before TDM overwrites it
        if (wave == 0 && c + 2 < NC) {
            const int kn = (c + 2) * KC;
            tdm_load_tile(x + (size_t)r0 * K + kn, (c & 1) ? xoff1 : xoff0, K);
            tdm_load_tile(w + (size_t)c0 * K + kn, (c & 1) ? woff1 : woff0, K);
        }
    }

    const float bv0 = bias[c0 + cw + lc];
    const float bv1 = bias[c0 + cw + 16 + lc];

    // C/D layout: VGPR v <-> row (v + 8*half), col = lane&15
    #pragma unroll
    for (int v = 0; v < 8; ++v) {
        const int ra = r0 + rw + v + 8 * half;
        const int rb = ra + 16;
        out[(size_t)ra * M + c0 + cw + lc]      = acc00[v] + bv0;
        out[(size_t)ra * M + c0 + cw + 16 + lc] = acc01[v] + bv1;
        out[(size_t)rb * M + c0 + cw + lc]      = acc10[v] + bv0;
        out[(size_t)rb * M + c0 + cw + 16 + lc] = acc11[v] + bv1;
    }
}

extern "C" void kernel_launch(void* const* d_in, const int* in_sizes, int n_in,
                              void* d_out, int out_size, void* d_ws, size_t ws_size,
                              hipStream_t stream) {
    const float* x    = (const float*)d_in[0];   // [N, K]
    const float* wgt  = (const float*)d_in[1];   // [M, K]
    const float* bias = (const float*)d_in[2];   // [M]
    float* out        = (float*)d_out;           // [N, M]

    const int M = in_sizes[2];
    const int K = in_sizes[1] / M;
    const int N = in_sizes[0] / K;

    dim3 block(128);                 // 4 wave32s; each owns a 32x32 output tile
    dim3 grid(M / BT, N / BT);
    mitchell_linear_wmma_tdm<<<grid, block, 0, stream>>>(x, wgt, bias, out, K, M);
}